// ModelVLLM_28681791602768
// MI455X (gfx1250) — compile-verified
//
#include <hip/hip_runtime.h>

typedef float v2f __attribute__((ext_vector_type(2)));
typedef float v8f __attribute__((ext_vector_type(8)));

// Problem shape (fixed by the reference's setup_inputs).
constexpr int B_   = 2;
constexpr int H_   = 32;
constexpr int N_   = 8192;
constexpr int D_   = 128;   // d (K dimension)
constexpr int E_   = 128;   // e (N dimension)
constexpr int BLK  = 64;    // block size (M dimension per block)
constexpr int NBLK = N_ / BLK;           // 128 blocks per head

// LDS strides (floats), padded for conflict-free WMMA operand gathers.
constexpr int QS_STRIDE  = 68;    // Q:  64 K-chunk cols + 4 pad  (68 mod 64 = 4)
constexpr int KSP_STRIDE = 288;   // KV: 32 K-pair rows, 128 cols x2 interleaved + 32 pad
                                  //     (288 mod 64 = 32 -> lane halves in disjoint banks)

__global__ __launch_bounds__(256)
void lightning_inter_kernel(const float* __restrict__ q,
                            const float* __restrict__ out_in,
                            const float* __restrict__ s,
                            const float* __restrict__ kv,
                            float* __restrict__ out)
{
    __shared__ float qs[BLK * QS_STRIDE];   // 64 x 64 Q K-chunk (+pad)          17,408 B
    __shared__ float ks[32 * KSP_STRIDE];   // 32 K-pair rows x 128 cols x 2     36,864 B

    const int tid    = threadIdx.x;
    const int lane   = tid & 31;
    const int wv     = tid >> 5;       // wave 0..7
    const int laneLo = lane & 15;      // 0..15
    const int hi     = lane >> 4;      // 0 or 1
    const int mt     = wv >> 1;        // M tile 0..3 (16 rows each)
    const int nh     = wv & 1;         // N half (0 -> cols 0..63, 1 -> 64..127)

    const int blk = blockIdx.x;        // 0 .. B*H*NBLK-1
    const int bh  = blk / NBLK;
    const int m   = blk - bh * NBLK;
    const int h   = bh & (H_ - 1);

    const size_t q_base  = ((size_t)bh * N_ + (size_t)m * BLK) * D_;
    const size_t o_base  = ((size_t)bh * N_ + (size_t)m * BLK) * E_;
    const size_t kv_base = ((size_t)bh * NBLK + (size_t)m) * (size_t)D_ * E_;

    v8f acc[4] = {};   // four 16x16 f32 C/D tiles (8 VGPRs each, per ISA layout)

    // Two K-phases of 64 each (keeps LDS under 64KB).
    for (int kc = 0; kc < 2; ++kc) {
        // ---- Q staging: async global->LDS b128 (ASYNCcnt path), 4 per thread ----
        // 64 rows x 64 cols = 1024 float4.
        #pragma unroll
        for (int i = 0; i < 4; ++i) {
            int idx = tid + i * 256;               // 0..1023
            int row = idx >> 4;                    // 16 float4 per row
            int c4  = idx & 15;
            const float* gsrc = q + q_base + (size_t)row * D_ + (size_t)(kc * 64 + c4 * 4);
            unsigned long long gaddr = (unsigned long long)(uintptr_t)gsrc;
            // Generic pointer to LDS: low 32 bits are the LDS byte offset.
            unsigned ldsoff = (unsigned)(uintptr_t)(&qs[row * QS_STRIDE + c4 * 4]);
            asm volatile("global_load_async_to_lds_b128 %0, %1, off"
                         :: "v"(ldsoff), "v"(gaddr)
                         : "memory");
        }

        // ---- KV staging: K-pair interleave in registers, b128 in / b128 out ----
        // Unit = (pair-row p, 4 cols): rows 2p,2p+1 -> 8 contiguous interleaved floats.
        // 32 pair-rows x 32 col-groups = 1024 units, 4 per thread.
        #pragma unroll
        for (int i = 0; i < 4; ++i) {
            int u  = tid + i * 256;                // 0..1023
            int c4 = u & 31;                       // col group (cols 4*c4 .. 4*c4+3)
            int p  = u >> 5;                       // pair-row 0..31 (K rows 2p,2p+1)
            const float* g0 = kv + kv_base + (size_t)(kc * 64 + 2 * p) * E_ + (size_t)(c4 * 4);
            float4 a = *reinterpret_cast<const float4*>(g0);        // K = 2p
            float4 b = *reinterpret_cast<const float4*>(g0 + E_);   // K = 2p+1
            float* dst = &ks[p * KSP_STRIDE + c4 * 8];
            float4 lo = make_float4(a.x, b.x, a.y, b.y);
            float4 hs = make_float4(a.z, b.z, a.w, b.w);
            reinterpret_cast<float4*>(dst)[0] = lo;
            reinterpret_cast<float4*>(dst)[1] = hs;
        }

        asm volatile("s_wait_asynccnt 0x0" ::: "memory");
        __syncthreads();

        // ---- WMMA main loop: 16 K-steps of 4 over this chunk ----
        #pragma unroll 4
        for (int kk = 0; kk < 16; ++kk) {
            const int k0 = kk * 4;
            // A tile 16x4: lane = M row (laneLo), VGPR pair holds K = k0+2*hi, +1
            v2f a = *reinterpret_cast<const v2f*>(
                &qs[(mt * 16 + laneLo) * QS_STRIDE + k0 + 2 * hi]);
            #pragma unroll
            for (int j = 0; j < 4; ++j) {
                const int nb = (nh * 4 + j) * 16;  // N tile base column
                // B tile 4x16: lane = N col; interleaved pair (K=k0+2hi, k0+2hi+1)
                // is contiguous -> single ds_load_b64, no re-pairing movs.
                v2f bv = *reinterpret_cast<const v2f*>(
                    &ks[(2 * kk + hi) * KSP_STRIDE + (nb + laneLo) * 2]);
                acc[j] = __builtin_amdgcn_wmma_f32_16x16x4_f32(
                    /*neg_a=*/false, a, /*neg_b=*/false, bv,
                    /*c_mod=*/(short)0, acc[j],
                    /*reuse_a=*/false, /*reuse_b=*/false);
            }
        }
        __syncthreads();
    }

    // ---- Epilogue: per-row decay, out = out_in + decay * acc ----
    const float sh = s[h];
    float dec[8];
    #pragma unroll
    for (int r = 0; r < 8; ++r) {
        const int row = mt * 16 + hi * 8 + r;      // row within 64-block (C/D layout)
        dec[r] = __expf(-sh * (float)(row + 1));
    }

    #pragma unroll
    for (int j = 0; j < 4; ++j) {
        const int nb = (nh * 4 + j) * 16;
        #pragma unroll
        for (int r = 0; r < 8; ++r) {
            const int row = mt * 16 + hi * 8 + r;
            const size_t idx = o_base + (size_t)row * E_ + (size_t)(nb + laneLo);
            out[idx] = out_in[idx] + dec[r] * acc[j][r];
        }
    }
}

extern "C" void kernel_launch(void* const* d_in, const int* in_sizes, int n_in,
                              void* d_out, int out_size, void* d_ws, size_t ws_size,
                              hipStream_t stream) {
    (void)in_sizes; (void)n_in; (void)out_size; (void)d_ws; (void)ws_size;
    const float* q   = (const float*)d_in[0];   // query  [b,h,n,d]
    const float* oin = (const float*)d_in[1];   // output (pre-filled diag part) [b,h,n,e]
    const float* s   = (const float*)d_in[2];   // decay slopes [h]
    const float* kv  = (const float*)d_in[3];   // kv [b,h,num_block,d,e]
    float* out = (float*)d_out;

    dim3 grid(B_ * H_ * NBLK);   // 8192 workgroups, one per (b,h,block)
    dim3 block(256);             // 8 waves (wave32)
    lightning_inter_kernel<<<grid, block, 0, stream>>>(q, oin, s, kv, out);
}